// MultiRelationalGATConv_18090402250932
// MI455X (gfx1250) — compile-verified
//
#include <hip/hip_runtime.h>
#include <hip/hip_bf16.h>

#define GAT_N 100000
#define GAT_E 600000
#define GAT_D 128
#define NEG_SLOPE 0.2f

typedef __attribute__((ext_vector_type(2))) float v2f;
typedef __attribute__((ext_vector_type(8))) float v8f;

// Monotone float<->int encoding for atomicMax on signed int (involution).
__device__ __forceinline__ int ordered_enc(float f) {
    int i = __float_as_int(f);
    return (i >= 0) ? i : (i ^ 0x7FFFFFFF);
}
__device__ __forceinline__ float ordered_dec(int i) {
    return __int_as_float((i >= 0) ? i : (i ^ 0x7FFFFFFF));
}

// ---------------------------------------------------------------------------
// h = x @ W   (N x 128) @ (128 x 128), fp32 via V_WMMA_F32_16X16X4_F32.
// One wave32 per 16x16 output tile; K loop of 32 WMMAs (K=4 each).
// ---------------------------------------------------------------------------
__global__ __launch_bounds__(256) void gemm_xW(const float* __restrict__ x,
                                               const float* __restrict__ W,
                                               float* __restrict__ h) {
    const int wave  = (blockIdx.x * 256 + threadIdx.x) >> 5;   // global wave id
    const int lane  = threadIdx.x & 31;
    const int tileM = (wave >> 3) * 16;      // 6250 row tiles (N divisible by 16)
    const int tileN = (wave & 7) * 16;       // 8 col tiles
    const int half  = lane >> 4;             // 0: lanes 0-15, 1: lanes 16-31
    const int l16   = lane & 15;

    const float* __restrict__ xrow = x + (size_t)(tileM + l16) * GAT_D;

    v8f c = {};
#pragma unroll 4
    for (int k = 0; k < GAT_D; k += 4) {
        // A 16x4 f32: lanes 0-15 hold K={k,k+1}, lanes 16-31 hold K={k+2,k+3}
        const float2 ax = *(const float2*)(xrow + k + 2 * half);
        v2f a; a.x = ax.x; a.y = ax.y;
        // B 4x16 f32: VGPR0 = row k (lanes 0-15) / row k+2 (lanes 16-31); VGPR1 = k+1 / k+3
        v2f b;
        b.x = W[(size_t)(k + 2 * half)     * GAT_D + tileN + l16];
        b.y = W[(size_t)(k + 2 * half + 1) * GAT_D + tileN + l16];
        c = __builtin_amdgcn_wmma_f32_16x16x4_f32(false, a, false, b,
                                                  (short)0, c, false, false);
    }

    // C/D layout: VGPR v -> row tileM + v + 8*half, col tileN + l16
    float* __restrict__ hout = h + (size_t)tileM * GAT_D + tileN;
#pragma unroll
    for (int v = 0; v < 8; ++v) {
        hout[(size_t)(v + 8 * half) * GAT_D + l16] = c[v];
    }
}

// ---------------------------------------------------------------------------
// Per-node attention scalars: s1[n] = h[n] . a[0:128], s2[n] = h[n] . a[128:256]
// One wave32 per node, float4 loads (32 lanes * 4 = 128), shuffle reduction.
// ---------------------------------------------------------------------------
__global__ __launch_bounds__(256) void node_scores(const float* __restrict__ h,
                                                   const float* __restrict__ avec,
                                                   float* __restrict__ s1,
                                                   float* __restrict__ s2) {
    const int node = (blockIdx.x * 256 + threadIdx.x) >> 5;
    const int lane = threadIdx.x & 31;
    if (node >= GAT_N) return;

    const float4 hv  = ((const float4*)(h + (size_t)node * GAT_D))[lane];
    const float4 av1 = ((const float4*)(avec))[lane];
    const float4 av2 = ((const float4*)(avec + GAT_D))[lane];

    float d1 = hv.x * av1.x + hv.y * av1.y + hv.z * av1.z + hv.w * av1.w;
    float d2 = hv.x * av2.x + hv.y * av2.y + hv.z * av2.z + hv.w * av2.w;
#pragma unroll
    for (int off = 16; off > 0; off >>= 1) {
        d1 += __shfl_down(d1, off, 32);
        d2 += __shfl_down(d2, off, 32);
    }
    if (lane == 0) { s1[node] = d1; s2[node] = d2; }
}

// ---------------------------------------------------------------------------
// Init / small kernels
// ---------------------------------------------------------------------------
__global__ void init_out_bias(float* __restrict__ out, const float* __restrict__ bias) {
    int i = blockIdx.x * blockDim.x + threadIdx.x;
    if (i < GAT_N * GAT_D) out[i] = bias[i & (GAT_D - 1)];
}

__global__ void init_node_accum(int* __restrict__ emax, float* __restrict__ denom) {
    int i = blockIdx.x * blockDim.x + threadIdx.x;
    if (i < GAT_N) {
        emax[i] = ordered_enc(-__builtin_inff());
        denom[i] = 0.0f;
    }
}

__global__ void softmax3(const float* __restrict__ logits, float* __restrict__ w) {
    if (threadIdx.x == 0 && blockIdx.x == 0) {
        float m = fmaxf(logits[0], fmaxf(logits[1], logits[2]));
        float e0 = __expf(logits[0] - m);
        float e1 = __expf(logits[1] - m);
        float e2 = __expf(logits[2] - m);
        float s = e0 + e1 + e2;
        w[0] = e0 / s; w[1] = e1 / s; w[2] = e2 / s;
    }
}

// ---------------------------------------------------------------------------
// Edge phase 1: logits + leaky-relu + segment-max (ordered-int atomicMax)
// ---------------------------------------------------------------------------
__global__ void edge_logits(const int* __restrict__ src, const int* __restrict__ dst,
                            const float* __restrict__ ea,
                            const float* __restrict__ s1, const float* __restrict__ s2,
                            const float* __restrict__ avec,
                            float* __restrict__ e, int* __restrict__ emax) {
    int j = blockIdx.x * blockDim.x + threadIdx.x;
    if (j >= GAT_E) return;
    int d = dst[j];
    float v = s1[src[j]] + s2[d] + ea[j] * avec[2 * GAT_D];
    v = (v > 0.0f) ? v : NEG_SLOPE * v;
    e[j] = v;
    atomicMax(&emax[d], ordered_enc(v));
}

// ---------------------------------------------------------------------------
// Edge phase 2: alpha = exp(e - emax[dst]) (stored in-place), segment-sum denom
// ---------------------------------------------------------------------------
__global__ void edge_alpha(const int* __restrict__ dst,
                           float* __restrict__ e,        // in: logits, out: alpha
                           const int* __restrict__ emax,
                           float* __restrict__ denom) {
    int j = blockIdx.x * blockDim.x + threadIdx.x;
    if (j >= GAT_E) return;
    int d = dst[j];
    float al = __expf(e[j] - ordered_dec(emax[d]));
    e[j] = al;
    atomicAdd(&denom[d], al);
}

// ---------------------------------------------------------------------------
// Edge phase 3: out[dst] += w_r * (alpha/denom) * max(|ea|,0.01) * h[src]
// One wave32 per edge; float4 gather of h[src], 4 f32 atomics per lane.
// ---------------------------------------------------------------------------
__global__ __launch_bounds__(256) void edge_msg(const int* __restrict__ src,
                                                const int* __restrict__ dst,
                                                const float* __restrict__ ea,
                                                const float* __restrict__ alpha,
                                                const float* __restrict__ denom,
                                                const float* __restrict__ h,
                                                const float* __restrict__ wrel,
                                                float* __restrict__ out) {
    const int j    = (blockIdx.x * 256 + threadIdx.x) >> 5;
    const int lane = threadIdx.x & 31;
    if (j >= GAT_E) return;
    const int s = src[j];
    const int d = dst[j];
    const float coef = (alpha[j] / fmaxf(denom[d], 1e-12f))
                     * fmaxf(fabsf(ea[j]), 0.01f) * wrel[0];
    const float4 m = ((const float4*)(h + (size_t)s * GAT_D))[lane];
    float* o = out + (size_t)d * GAT_D + lane * 4;
    atomicAdd(o + 0, coef * m.x);
    atomicAdd(o + 1, coef * m.y);
    atomicAdd(o + 2, coef * m.z);
    atomicAdd(o + 3, coef * m.w);
}

// ---------------------------------------------------------------------------
// Host orchestration
// ---------------------------------------------------------------------------
extern "C" void kernel_launch(void* const* d_in, const int* in_sizes, int n_in,
                              void* d_out, int out_size, void* d_ws, size_t ws_size,
                              hipStream_t stream) {
    (void)in_sizes; (void)n_in; (void)out_size; (void)ws_size;

    const float* x     = (const float*)d_in[0];
    const int*   eidx[3] = { (const int*)d_in[1], (const int*)d_in[2], (const int*)d_in[3] };
    const float* ea[3]   = { (const float*)d_in[4], (const float*)d_in[5], (const float*)d_in[6] };
    const float* Wm[3]   = { (const float*)d_in[7], (const float*)d_in[8], (const float*)d_in[9] };
    const float* av[3]   = { (const float*)d_in[10], (const float*)d_in[11], (const float*)d_in[12] };
    const float* rlog  = (const float*)d_in[13];
    const float* bias  = (const float*)d_in[14];
    float* out = (float*)d_out;

    // Workspace layout (floats), relation-sequential reuse. ~55 MB total.
    float* h     = (float*)d_ws;                     // N*128
    float* s1    = h + (size_t)GAT_N * GAT_D;        // N
    float* s2    = s1 + GAT_N;                       // N
    float* e     = s2 + GAT_N;                       // E (logits, then alpha in-place)
    int*   emax  = (int*)(e + GAT_E);                // N
    float* denom = (float*)(emax + GAT_N);           // N
    float* w3    = denom + GAT_N;                    // 3

    softmax3<<<1, 32, 0, stream>>>(rlog, w3);
    init_out_bias<<<(GAT_N * GAT_D + 255) / 256, 256, 0, stream>>>(out, bias);

    const int gemm_blocks  = (GAT_N / 16) * 8 / 8;          // 6250 (8 waves/block)
    const int score_blocks = (GAT_N * 32 + 255) / 256;      // 1 wave per node
    const int edge_blocks  = (GAT_E + 255) / 256;
    const int msg_blocks   = (GAT_E * 32 + 255) / 256;      // 1 wave per edge

    for (int r = 0; r < 3; ++r) {
        const int* src = eidx[r];
        const int* dst = eidx[r] + GAT_E;

        gemm_xW<<<gemm_blocks, 256, 0, stream>>>(x, Wm[r], h);
        node_scores<<<score_blocks, 256, 0, stream>>>(h, av[r], s1, s2);
        init_node_accum<<<(GAT_N + 255) / 256, 256, 0, stream>>>(emax, denom);
        edge_logits<<<edge_blocks, 256, 0, stream>>>(src, dst, ea[r], s1, s2, av[r], e, emax);
        edge_alpha<<<edge_blocks, 256, 0, stream>>>(dst, e, emax, denom);
        edge_msg<<<msg_blocks, 256, 0, stream>>>(src, dst, ea[r], e, denom, h, w3 + r, out);
    }
}